// GraphEncoder_31636729102882
// MI455X (gfx1250) — compile-verified
//
#include <hip/hip_runtime.h>

typedef __attribute__((ext_vector_type(2))) float v2f;
typedef __attribute__((ext_vector_type(8))) float v8f;

#define N_NODES 20000
#define N_EDGES 320000
#define IN_DIM 768
#define HID 256
#define HEADS 4
#define CH 64
#define NEG_SLOPE 0.2f
#define E_TOT (N_EDGES + N_NODES)   // edges + self loops

// ---------------------------------------------------------------------------
// GEMM: C[M,N] = A[M,K] @ B[K,N], fp32, via V_WMMA_F32_16X16X4_F32.
// One wave32 per 16x16 output tile. M,N multiples of 16; K multiple of 4.
// A 16x4 layout: lanes 0-15 -> M=lane, VGPR0/1 = K={0,1}; lanes 16-31 -> K={2,3}.
// B 4x16 layout mirrors A. C/D: VGPR r -> row r (lanes 0-15) / row r+8 (16-31).
// ---------------------------------------------------------------------------
__global__ __launch_bounds__(256) void gemm_wmma_f32(
    const float* __restrict__ A, const float* __restrict__ B,
    float* __restrict__ C, int M, int N, int K)
{
    const int wave   = (blockIdx.x * blockDim.x + threadIdx.x) >> 5;  // uniform in wave
    const int lane   = threadIdx.x & 31;
    const int ntn    = N >> 4;
    const int tile_m = wave / ntn;
    const int tile_n = wave % ntn;
    if (tile_m * 16 >= M) return;   // uniform per wave; never taken for exact grids

    const int half = lane >> 4;     // 0: K pair {0,1}; 1: K pair {2,3}
    const int l    = lane & 15;

    const float* __restrict__ Arow = A + (size_t)(tile_m * 16 + l) * K;
    const float* __restrict__ Bcol = B + (size_t)(tile_n * 16 + l);

    v8f acc = {};
    for (int k = 0; k < K; k += 4) {
        v2f a, b;
        const int kk = k + half * 2;
        a.x = Arow[kk + 0];
        a.y = Arow[kk + 1];
        b.x = Bcol[(size_t)(kk + 0) * N];
        b.y = Bcol[(size_t)(kk + 1) * N];
        acc = __builtin_amdgcn_wmma_f32_16x16x4_f32(
            /*neg_a=*/false, a, /*neg_b=*/false, b,
            /*c_mod=*/(short)0, acc, /*reuse_a=*/false, /*reuse_b=*/false);
    }

    float* __restrict__ Cb = C + (size_t)(tile_m * 16) * N + tile_n * 16;
#pragma unroll
    for (int r = 0; r < 8; ++r) {
        Cb[(size_t)(half * 8 + r) * N + l] = acc[r];
    }
}

// ---------------------------------------------------------------------------
// Per-node attention logits: al_s[n,h] = <h[n,h,:], a_src[h,:]>, same for dst.
// ---------------------------------------------------------------------------
__global__ void node_logits(const float* __restrict__ h,
                            const float* __restrict__ a_src,
                            const float* __restrict__ a_dst,
                            float* __restrict__ al_s,
                            float* __restrict__ al_d)
{
    const int idx = blockIdx.x * blockDim.x + threadIdx.x;   // n*HEADS + hd
    if (idx >= N_NODES * HEADS) return;
    const int n  = idx / HEADS;
    const int hd = idx % HEADS;
    const float* __restrict__ hp = h + (size_t)n * HID + hd * CH;
    const float* __restrict__ as = a_src + hd * CH;
    const float* __restrict__ ad = a_dst + hd * CH;
    float ss = 0.f, sd = 0.f;
#pragma unroll 4
    for (int c = 0; c < CH; ++c) {
        const float v = hp[c];
        ss += v * as[c];
        sd += v * ad[c];
    }
    al_s[idx] = ss;
    al_d[idx] = sd;
}

// ---------------------------------------------------------------------------
// Init: m = -inf, s = 0, acc = 0. Grid covers N_NODES*HID.
// ---------------------------------------------------------------------------
__global__ void init_buffers(float* __restrict__ m, float* __restrict__ s,
                             float* __restrict__ acc)
{
    const int i = blockIdx.x * blockDim.x + threadIdx.x;
    if (i < N_NODES * HEADS) {
        m[i] = -__builtin_inff();
        s[i] = 0.f;
    }
    if (i < N_NODES * HID) acc[i] = 0.f;
}

// Float atomic-max via signed max / unsigned min on the raw bits.
__device__ inline void atomicMaxFloat(float* addr, float val)
{
    if (val >= 0.f) atomicMax((int*)addr, __float_as_int(val));
    else            atomicMin((unsigned int*)addr, __float_as_uint(val));
}

__device__ inline float leaky(float x) { return x > 0.f ? x : x * NEG_SLOPE; }

// ---------------------------------------------------------------------------
// Pass 1: segment max of leaky-relu edge logits into m[dst,h].
// One thread per (edge, head); edge >= N_EDGES means self loop.
// ---------------------------------------------------------------------------
__global__ void edge_max(const int* __restrict__ esrc, const int* __restrict__ edst,
                         const float* __restrict__ al_s, const float* __restrict__ al_d,
                         float* __restrict__ m)
{
    const int idx = blockIdx.x * blockDim.x + threadIdx.x;   // e*HEADS + hd
    if (idx >= E_TOT * HEADS) return;
    const int e  = idx / HEADS;
    const int hd = idx % HEADS;
    int s, d;
    if (e < N_EDGES) { s = esrc[e]; d = edst[e]; }
    else             { s = d = e - N_EDGES; }
    const float ev = leaky(al_s[s * HEADS + hd] + al_d[d * HEADS + hd]);
    atomicMaxFloat(&m[d * HEADS + hd], ev);
}

// ---------------------------------------------------------------------------
// Pass 2: p = exp(e - m[dst]); s[dst,h] += p; acc[dst,h,c] += p * h[src,h,c].
// One wave32 per edge; lanes cover channels (c = lane, lane+32) per head.
// ---------------------------------------------------------------------------
__global__ __launch_bounds__(256) void edge_acc(
    const int* __restrict__ esrc, const int* __restrict__ edst,
    const float* __restrict__ al_s, const float* __restrict__ al_d,
    const float* __restrict__ m, const float* __restrict__ h,
    float* __restrict__ sden, float* __restrict__ acc)
{
    const int wave = (blockIdx.x * blockDim.x + threadIdx.x) >> 5;
    const int lane = threadIdx.x & 31;
    if (wave >= E_TOT) return;
    int s, d;
    if (wave < N_EDGES) { s = esrc[wave]; d = edst[wave]; }
    else                { s = d = wave - N_EDGES; }

    const float* __restrict__ hs = h   + (size_t)s * HID;
    float*       __restrict__ ac = acc + (size_t)d * HID;

#pragma unroll
    for (int hd = 0; hd < HEADS; ++hd) {
        const float ev = leaky(al_s[s * HEADS + hd] + al_d[d * HEADS + hd]);
        const float p  = __expf(ev - m[d * HEADS + hd]);
        if (lane == 0) atomicAdd(&sden[d * HEADS + hd], p);
        const int base = hd * CH;
        atomicAdd(&ac[base + lane],      p * hs[base + lane]);
        atomicAdd(&ac[base + lane + 32], p * hs[base + lane + 32]);
    }
}

// ---------------------------------------------------------------------------
// Finalize: out[n, h*CH + c] = acc / (s + 1e-16) + bias.
// ---------------------------------------------------------------------------
__global__ void finalize(const float* __restrict__ acc, const float* __restrict__ sden,
                         const float* __restrict__ bias, float* __restrict__ out)
{
    const int idx = blockIdx.x * blockDim.x + threadIdx.x;
    if (idx >= N_NODES * HID) return;
    const int n   = idx / HID;
    const int col = idx % HID;
    const int hd  = col / CH;
    out[idx] = acc[idx] / (sden[n * HEADS + hd] + 1e-16f) + bias[col];
}

// ---------------------------------------------------------------------------
// Launch
// ---------------------------------------------------------------------------
extern "C" void kernel_launch(void* const* d_in, const int* in_sizes, int n_in,
                              void* d_out, int out_size, void* d_ws, size_t ws_size,
                              hipStream_t stream)
{
    const float* x      = (const float*)d_in[0];
    const int*   esrc   = (const int*)  d_in[1];
    const int*   edst   = (const int*)  d_in[2];
    const float* W1     = (const float*)d_in[3];
    const float* a1_src = (const float*)d_in[4];
    const float* a1_dst = (const float*)d_in[5];
    const float* b1     = (const float*)d_in[6];
    const float* W2     = (const float*)d_in[7];
    const float* a2_src = (const float*)d_in[8];
    const float* a2_dst = (const float*)d_in[9];
    const float* b2     = (const float*)d_in[10];
    float* out = (float*)d_out;

    // Workspace layout (floats)
    float* ws  = (float*)d_ws;
    const size_t NH = (size_t)N_NODES * HID;     // 5.12M
    const size_t NA = (size_t)N_NODES * HEADS;   // 80K
    float* h    = ws;                 // [N, HID]
    float* x1   = h    + NH;          // [N, HID]
    float* acc  = x1   + NH;          // [N, HID]
    float* al_s = acc  + NH;          // [N, HEADS]
    float* al_d = al_s + NA;
    float* mmax = al_d + NA;
    float* sden = mmax + NA;

    const int gemm_waves1 = (N_NODES / 16) * (HID / 16);     // 20000 waves
    const dim3 gemm_grid((gemm_waves1 * 32 + 255) / 256);
    const dim3 blk(256);
    const dim3 nl_grid((N_NODES * HEADS + 255) / 256);
    const dim3 init_grid((N_NODES * HID + 255) / 256);
    const dim3 emax_grid((E_TOT * HEADS + 255) / 256);
    const dim3 eacc_grid((E_TOT * 32 + 255) / 256);
    const dim3 fin_grid((N_NODES * HID + 255) / 256);

    // ------------------- Layer 1 -------------------
    gemm_wmma_f32<<<gemm_grid, blk, 0, stream>>>(x, W1, h, N_NODES, HID, IN_DIM);
    node_logits<<<nl_grid, blk, 0, stream>>>(h, a1_src, a1_dst, al_s, al_d);
    init_buffers<<<init_grid, blk, 0, stream>>>(mmax, sden, acc);
    edge_max<<<emax_grid, blk, 0, stream>>>(esrc, edst, al_s, al_d, mmax);
    edge_acc<<<eacc_grid, blk, 0, stream>>>(esrc, edst, al_s, al_d, mmax, h, sden, acc);
    finalize<<<fin_grid, blk, 0, stream>>>(acc, sden, b1, x1);

    // ------------------- Layer 2 -------------------
    gemm_wmma_f32<<<gemm_grid, blk, 0, stream>>>(x1, W2, h, N_NODES, HID, HID);
    node_logits<<<nl_grid, blk, 0, stream>>>(h, a2_src, a2_dst, al_s, al_d);
    init_buffers<<<init_grid, blk, 0, stream>>>(mmax, sden, acc);
    edge_max<<<emax_grid, blk, 0, stream>>>(esrc, edst, al_s, al_d, mmax);
    edge_acc<<<eacc_grid, blk, 0, stream>>>(esrc, edst, al_s, al_d, mmax, h, sden, acc);
    finalize<<<fin_grid, blk, 0, stream>>>(acc, sden, b2, out);
}